// SamplePatches_57148834841007
// MI455X (gfx1250) — compile-verified
//
#include <hip/hip_runtime.h>

// ---------------- problem constants (from reference) ----------------
#define BATCH   8
#define HL      128          // x_low H=W
#define HH      1024         // x_high H=W
#define CCH     3            // channels
#define AH      120          // attention H
#define AW      120          // attention W
#define AHW     (AH*AW)      // 14400
#define NP      10           // N_PATCHES
#define PH      100
#define PW      100
#define RF      8            // receptive field
#define ROW_F   (PW*CCH)     // 300 floats per patch row (contiguous in x_high)
#define SRC_ROW_F (HH*CCH)   // 3072 floats per x_high row
#define PATCH_F (PH*ROW_F)   // 30000 floats per patch
#define CHUNK_ROWS 25
#define CHUNK_F (CHUNK_ROWS*ROW_F)   // 7500 dwords -> 30KB LDS stage
#define NCHUNK  (PH/CHUNK_ROWS)      // 4

#define AS1 __attribute__((address_space(1)))
#define AS3 __attribute__((address_space(3)))

typedef unsigned int u32x4 __attribute__((ext_vector_type(4)));
typedef int          i32x8 __attribute__((ext_vector_type(8)));
typedef int          i32x4 __attribute__((ext_vector_type(4)));

#if __has_builtin(__builtin_amdgcn_tensor_load_to_lds) && \
    __has_builtin(__builtin_amdgcn_tensor_store_from_lds)
#define HAVE_TDM 1
#else
#define HAVE_TDM 0
#endif

// ---------------- CDNA5 async global<->LDS helpers (fallback path) ----------------
__device__ __forceinline__ void async_g2l_b32(const float* g, AS3 float* l) {
#if __has_builtin(__builtin_amdgcn_global_load_async_to_lds_b32)
  __builtin_amdgcn_global_load_async_to_lds_b32((AS1 int*)g, (AS3 int*)l, 0, 0);
#else
  asm volatile("global_load_async_to_lds_b32 %0, %1, off"
               :: "v"((unsigned)(unsigned long long)l),
                  "v"((unsigned long long)g)
               : "memory");
#endif
}

__device__ __forceinline__ void async_l2g_b32(float* g, AS3 float* l) {
#if __has_builtin(__builtin_amdgcn_global_store_async_from_lds_b32)
  __builtin_amdgcn_global_store_async_from_lds_b32((AS1 int*)g, (AS3 int*)l, 0, 0);
#else
  asm volatile("global_store_async_from_lds_b32 %0, %1, off"
               :: "v"((unsigned long long)g),
                  "v"((unsigned)(unsigned long long)l)
               : "memory");
#endif
}

__device__ __forceinline__ void wait_async0() {
#if __has_builtin(__builtin_amdgcn_s_wait_asynccnt)
  __builtin_amdgcn_s_wait_asynccnt(0);
#else
  asm volatile("s_wait_asynccnt 0" ::: "memory");
#endif
}

__device__ __forceinline__ void wait_tensor0() {
#if __has_builtin(__builtin_amdgcn_s_wait_tensorcnt)
  __builtin_amdgcn_s_wait_tensorcnt(0);
#else
  asm volatile("s_wait_tensorcnt 0" ::: "memory");
#endif
}

// Build D# group0: {flags, lds_addr, global_addr lo, global_addr hi | type=2}
__device__ __forceinline__ u32x4 tdm_group0(unsigned lds_addr, unsigned long long gaddr) {
  u32x4 g0;
  g0.x = 1u;                                     // count=1 valid, no gather, no restore
  g0.y = lds_addr;                               // LDS byte address
  g0.z = (unsigned)(gaddr & 0xFFFFFFFFull);      // global_addr[31:0]
  g0.w = (unsigned)((gaddr >> 32) & 0x01FFFFFFu) // global_addr[56:32]
         | 0x80000000u;                          // type=2 ("image") in bits[127:126]
  return g0;
}

// Build D# group1 for a 2D tile: tile_dim0 x tile_dim1 elements of 4B,
// row stride = stride0 elements. tensor_dims set huge (tiles pre-clamped in-bounds).
__device__ __forceinline__ i32x8 tdm_group1(unsigned tile_d0, unsigned tile_d1,
                                            unsigned stride0) {
  const unsigned dim0 = 0x7FFFFFFFu, dim1 = 0x7FFFFFFFu;
  i32x8 g1;
  g1[0] = (int)(2u << 16);                                   // data_size=2 -> 4 bytes
  g1[1] = (int)((dim0 & 0xFFFFu) << 16);                     // tensor_dim0[15:0]
  g1[2] = (int)(((dim1 & 0xFFFFu) << 16) | (dim0 >> 16));    // dim1[15:0] | dim0[31:16]
  g1[3] = (int)((tile_d0 << 16) | (dim1 >> 16));             // tile_dim0 | dim1[31:16]
  g1[4] = (int)(tile_d1);                                    // tile_dim1 (tile_dim2=0)
  g1[5] = (int)stride0;                                      // tensor_dim0_stride[31:0]
  g1[6] = 0;                                                 // stride0 hi | stride1 lo
  g1[7] = 0;                                                 // stride1 hi
  return g1;
}

// ---------------- kernel 1: Gumbel top-k + metadata ----------------
// one block per batch; 256 threads (8 wave32)
__global__ void SamplePatches_topk_kernel(const float* __restrict__ att,
                                          const float* __restrict__ un,
                                          float* __restrict__ out_satt,
                                          float* __restrict__ out_off,
                                          float* __restrict__ out_samp,
                                          int* __restrict__ corners)
{
  const int b   = blockIdx.x;
  const int tid = threadIdx.x;
  __shared__ float rv[256];
  __shared__ int   ri[256];
  __shared__ int   chosen[NP];

  const float* a = att + (size_t)b * AHW;
  const float* u = un  + (size_t)b * AHW;

  for (int k = 0; k < NP; ++k) {
    float bv = -__builtin_inff();
    int   bi = 0x7fffffff;
    for (int i = tid; i < AHW; i += 256) {
      bool skip = false;
      for (int j = 0; j < k; ++j) skip |= (chosen[j] == i);
      if (skip) continue;
      float uc  = fminf(fmaxf(u[i], 1e-6f), 1.0f - 1e-6f);
      float gum = -logf(-logf(uc));
      float key = logf(fmaxf(a[i], 1e-20f)) + gum;
      if (key > bv || (key == bv && i < bi)) { bv = key; bi = i; }
    }
    rv[tid] = bv; ri[tid] = bi;
    __syncthreads();
    for (int s = 128; s > 0; s >>= 1) {
      if (tid < s) {
        float ov = rv[tid + s]; int oi = ri[tid + s];
        if (ov > rv[tid] || (ov == rv[tid] && oi < ri[tid])) { rv[tid] = ov; ri[tid] = oi; }
      }
      __syncthreads();
    }
    if (tid == 0) {
      const int sel = ri[0];
      chosen[k] = sel;
      const int sy = sel / AW, sx = sel % AW;
      const float scale = (float)HH / (float)HL;             // low->high res
      const float stepy = (float)(HL - RF) / (float)AH;      // prev_patch / space
      const float stepx = (float)(HL - RF) / (float)AW;
      const float offy = (sy * stepy + RF * 0.5f) * scale;
      const float offx = (sx * stepx + RF * 0.5f) * scale;
      const int idx = b * NP + k;
      out_satt[idx]        = a[sel];
      out_off[idx*2 + 0]   = offy;
      out_off[idx*2 + 1]   = offx;
      out_samp[idx*2 + 0]  = (float)sy;
      out_samp[idx*2 + 1]  = (float)sx;
      int cy = (int)floorf(offy);
      int cx = (int)floorf(offx);
      cy = cy < 0 ? 0 : (cy > HH - PH ? HH - PH : cy);       // dynamic_slice clamp
      cx = cx < 0 ? 0 : (cx > HH - PW ? HH - PW : cx);
      corners[idx*2 + 0] = cy;
      corners[idx*2 + 1] = cx;
    }
    __syncthreads();
  }
}

// ---------------- kernel 2: patch gather via TDM (one wave32 per patch) ----------------
// TDM load: 25x300 f32 tile, src row stride 3072 elems -> LDS (30KB)
// TDM store: 25x300 f32 tile, dst row stride 300 elems (contiguous patch)
__global__ void SamplePatches_copy_kernel(const float* __restrict__ xh,
                                          const int* __restrict__ corners,
                                          float* __restrict__ out)
{
  __shared__ float smem[CHUNK_F];
  const int p  = blockIdx.x;
  const int b  = p / NP;
  const int cy = __builtin_amdgcn_readfirstlane(corners[p*2 + 0]);
  const int cx = __builtin_amdgcn_readfirstlane(corners[p*2 + 1]);

  const float* src = xh + ((size_t)((size_t)b * HH + cy) * HH + cx) * CCH;
  float*       dst = out + (size_t)p * PATCH_F;
  AS3 float*   ls  = (AS3 float*)smem;
  const unsigned lds_addr = (unsigned)(unsigned long long)ls;

#if HAVE_TDM
  const i32x4 z4 = {0, 0, 0, 0};
  const i32x8 z8 = {0, 0, 0, 0, 0, 0, 0, 0};
  const i32x8 g1_load  = tdm_group1(ROW_F, CHUNK_ROWS, SRC_ROW_F);
  const i32x8 g1_store = tdm_group1(ROW_F, CHUNK_ROWS, ROW_F);
  for (int ch = 0; ch < NCHUNK; ++ch) {
    const unsigned long long gsrc =
        (unsigned long long)(src + (size_t)ch * CHUNK_ROWS * SRC_ROW_F);
    const unsigned long long gdst =
        (unsigned long long)(dst + (size_t)ch * CHUNK_F);
    __builtin_amdgcn_tensor_load_to_lds(tdm_group0(lds_addr, gsrc), g1_load,
                                        z4, z4, z8, 0);
    wait_tensor0();                       // tile landed in LDS
    __builtin_amdgcn_tensor_store_from_lds(tdm_group0(lds_addr, gdst), g1_store,
                                           z4, z4, z8, 0);
    wait_tensor0();                       // LDS drained before reuse
  }
#else
  // fallback: per-lane async global<->LDS DMA (ASYNCcnt-tracked)
  for (int ch = 0; ch < NCHUNK; ++ch) {
    const int base = ch * CHUNK_F;
    for (int d = threadIdx.x; d < CHUNK_F; d += 32) {
      const int g = base + d;
      const int r = g / ROW_F;
      const int c = g - r * ROW_F;
      async_g2l_b32(src + (size_t)r * SRC_ROW_F + c, ls + d);
    }
    wait_async0();
    for (int d = threadIdx.x; d < CHUNK_F; d += 32) {
      async_l2g_b32(dst + base + d, ls + d);
    }
    wait_async0();
  }
#endif
}

// ---------------- host launch ----------------
extern "C" void kernel_launch(void* const* d_in, const int* in_sizes, int n_in,
                              void* d_out, int out_size, void* d_ws, size_t ws_size,
                              hipStream_t stream) {
  (void)in_sizes; (void)n_in; (void)out_size; (void)ws_size;
  // inputs per setup_inputs(): x_low, x_high, attention, uniform_noise
  const float* x_high = (const float*)d_in[1];
  const float* att    = (const float*)d_in[2];
  const float* noise  = (const float*)d_in[3];

  float* out         = (float*)d_out;
  float* out_patches = out;                                  // (8,10,100,100,3)
  float* out_satt    = out + (size_t)BATCH * NP * PATCH_F;   // (8,10)
  float* out_off     = out_satt + BATCH * NP;                // (8,10,2)
  float* out_samp    = out_off + BATCH * NP * 2;             // (8,10,2)
  int*   corners     = (int*)d_ws;                           // (8,10,2) scratch

  SamplePatches_topk_kernel<<<BATCH, 256, 0, stream>>>(att, noise, out_satt,
                                                       out_off, out_samp, corners);
  SamplePatches_copy_kernel<<<BATCH * NP, 32, 0, stream>>>(x_high, corners, out_patches);
}